// AffinityPropogationUnit_41592463294565
// MI455X (gfx1250) — compile-verified
//
#include <hip/hip_runtime.h>

// N=8, C=64, H=W=256 -> HW=65536.
#define NB  8
#define C64 64
#define HWD 65536

typedef __attribute__((ext_vector_type(2))) float v2f;
typedef __attribute__((ext_vector_type(8))) float v8f;

// ---------------------------------------------------------------------------
// One wave computes a 64x16 strip of Y = W @ X + b  (K = 64) using
// V_WMMA_F32_16X16X4_F32:  D(16x16) = A(16x4) x B(4x16) + C, 16 k-steps.
//   M = output channel (4 tiles of 16), N = 16 spatial positions, K = in-chan.
// A-frag per lane (half=lane>>4, l=lane&15):  a.x=A[l, 2*half], a.y=A[l, 2*half+1]
// B-frag per lane:                            b.x=B[2*half, l], b.y=B[2*half+1, l]
// D readout: acc[r] on lane(half,l) = D[r + 8*half, l]
// WtL = weights transposed [c][o], LDS stride 65 (bank-conflict free).
// ---------------------------------------------------------------------------
__device__ __forceinline__ void gemm64_wave(
    const float* __restrict__ Xn,   // (64, HWD) input for image n
    const float* __restrict__ WtL,  // LDS transposed weights, stride 65
    const float* __restrict__ bL,   // LDS bias[64]
    int p0, int lane, v8f acc[4])
{
  const int half = lane >> 4;
  const int l    = lane & 15;
#pragma unroll
  for (int t = 0; t < 4; ++t)
#pragma unroll
    for (int r = 0; r < 8; ++r)
      acc[t][r] = bL[t * 16 + r + 8 * half];

#pragma unroll 4
  for (int kk = 0; kk < 16; ++kk) {
    const int cA = kk * 4 + 2 * half;   // this lane-half's K pair base
    v2f b;
    b.x = Xn[cA * HWD + p0 + l];
    b.y = Xn[(cA + 1) * HWD + p0 + l];
#pragma unroll
    for (int t = 0; t < 4; ++t) {
      v2f a;
      a.x = WtL[cA * 65 + t * 16 + l];
      a.y = WtL[(cA + 1) * 65 + t * 16 + l];
      acc[t] = __builtin_amdgcn_wmma_f32_16x16x4_f32(
          /*neg_a=*/false, a, /*neg_b=*/false, b,
          /*c_mod=*/(short)0, acc[t], /*reuse_a=*/false, /*reuse_b=*/false);
    }
  }
}

// Stage W (row-major [o][c]) transposed into LDS [c][o] (stride 65) + bias.
__device__ __forceinline__ void load_wt_bias(
    const float* __restrict__ Wg, const float* __restrict__ bg,
    float* WtL, float* bL, int tid, int nthr)
{
  for (int i = tid; i < 4096; i += nthr)
    WtL[(i & 63) * 65 + (i >> 6)] = Wg[i];   // coalesced read, conflict-free write
  if (tid < 64) bL[tid] = bg[tid];
}

// ---------------------------------------------------------------------------
// K1: outF = W_F @ seg + b_F  (written to workspace)
// grid (512, 8), 256 threads (8 waves x 16 positions = 128 positions/block)
// ---------------------------------------------------------------------------
__global__ void __launch_bounds__(256) k_convF(
    const float* __restrict__ seg, const float* __restrict__ WF,
    const float* __restrict__ bF, float* __restrict__ outF)
{
  __shared__ float WtL[64 * 65];
  __shared__ float bL[64];
  const int tid = threadIdx.x;
  load_wt_bias(WF, bF, WtL, bL, tid, 256);
  __syncthreads();

  const int n    = blockIdx.y;
  const int w    = tid >> 5, lane = tid & 31;
  const int half = lane >> 4, l = lane & 15;
  const int p0   = blockIdx.x * 128 + w * 16;
  const float* Xn = seg  + (size_t)n * C64 * HWD;
  float*       Yn = outF + (size_t)n * C64 * HWD;

  v8f acc[4];
  gemm64_wave(Xn, WtL, bL, p0, lane, acc);

#pragma unroll
  for (int t = 0; t < 4; ++t)
#pragma unroll
    for (int r = 0; r < 8; ++r)
      Yn[(size_t)(t * 16 + r + 8 * half) * HWD + p0 + l] = acc[t][r];
}

// ---------------------------------------------------------------------------
// K2: diag[n,x] = sum_k outF[n, x>>10, (x&1023)*64 + k] * outK[n, k, x]
// K-conv computed on the fly (waves 0-3, WMMA) while waves 4-7 stage the
// contiguous F segment. grid (1024, 8): 64 x's per block, channel-aligned.
// ---------------------------------------------------------------------------
__global__ void __launch_bounds__(256) k_diag(
    const float* __restrict__ seg, const float* __restrict__ WK,
    const float* __restrict__ bK, const float* __restrict__ outF,
    float* __restrict__ diag)
{
  __shared__ float WtL[64 * 65];
  __shared__ float bL[64];
  __shared__ float Ktile[64 * 64];   // [k][dx]
  __shared__ float Ftile[64 * 65];   // [dq][k], padded

  const int tid = threadIdx.x;
  load_wt_bias(WK, bK, WtL, bL, tid, 256);

  const int n  = blockIdx.y;
  const int x0 = blockIdx.x * 64;
  const int ch = x0 >> 10;           // F channel (uniform: 64 | 1024)
  const int q0 = x0 & 1023;          // q base within channel
  const float* Xn = seg  + (size_t)n * C64 * HWD;
  const float* Fn = outF + (size_t)n * C64 * HWD + (size_t)ch * HWD + (size_t)q0 * 64;
  __syncthreads();

  const int w = tid >> 5, lane = tid & 31;
  const int half = lane >> 4, l = lane & 15;

  if (w < 4) {                       // wave-uniform branch: EXEC all-1s inside
    v8f acc[4];
    gemm64_wave(Xn, WtL, bL, x0 + w * 16, lane, acc);
#pragma unroll
    for (int t = 0; t < 4; ++t)
#pragma unroll
      for (int r = 0; r < 8; ++r)
        Ktile[(t * 16 + r + 8 * half) * 64 + w * 16 + l] = acc[t][r];
  } else {
    for (int i = tid - 128; i < 4096; i += 128)
      Ftile[(i >> 6) * 65 + (i & 63)] = Fn[i];   // coalesced
  }
  __syncthreads();

  if (tid < 64) {                    // one x per thread: 64-term diagonal dot
    float s = 0.f;
#pragma unroll 8
    for (int k = 0; k < 64; ++k)
      s += Ftile[tid * 65 + k] * Ktile[k * 64 + tid];
    diag[(size_t)n * HWD + x0 + tid] = s;
  }
}

// ---------------------------------------------------------------------------
// K3/K5: outG = W_G @ depth + b_G; v = outG * diag[o*1024 + p>>6]
//   WRITE_OUT=false: accumulate per-channel sum / sumsq (BN stats)
//   WRITE_OUT=true : write gamma*(v-mean)*rsqrt(var+eps)+beta + depth
// ---------------------------------------------------------------------------
template <bool WRITE_OUT>
__global__ void __launch_bounds__(256) k_convG(
    const float* __restrict__ depth, const float* __restrict__ WG,
    const float* __restrict__ bG, const float* __restrict__ diag,
    const float* __restrict__ ss,                 // scale[64], shift[64]
    float* __restrict__ gsum, float* __restrict__ gsq,
    float* __restrict__ out)
{
  __shared__ float WtL[64 * 65];
  __shared__ float bL[64];
  __shared__ float lsum[64];
  __shared__ float lsq[64];

  const int tid = threadIdx.x;
  load_wt_bias(WG, bG, WtL, bL, tid, 256);
  if (!WRITE_OUT && tid < 64) { lsum[tid] = 0.f; lsq[tid] = 0.f; }
  __syncthreads();

  const int n    = blockIdx.y;
  const int w    = tid >> 5, lane = tid & 31;
  const int half = lane >> 4, l = lane & 15;
  const int p0   = blockIdx.x * 128 + w * 16;
  const int q    = p0 >> 6;          // wave-uniform: 16-run never crosses 64-group
  const float* Xn  = depth + (size_t)n * C64 * HWD;
  const float* dgn = diag  + (size_t)n * HWD;

  v8f acc[4];
  gemm64_wave(Xn, WtL, bL, p0, lane, acc);

#pragma unroll
  for (int t = 0; t < 4; ++t) {
#pragma unroll
    for (int r = 0; r < 8; ++r) {
      const int o = t * 16 + r + 8 * half;
      const float v = acc[t][r] * dgn[o * 1024 + q];
      if (WRITE_OUT) {
        out[(size_t)n * C64 * HWD + (size_t)o * HWD + p0 + l] =
            ss[o] * v + ss[64 + o] + Xn[(size_t)o * HWD + p0 + l];
      } else {
        float s1 = v, s2 = v * v;
#pragma unroll
        for (int off = 1; off < 16; off <<= 1) {  // reduce each 16-lane half
          s1 += __shfl_xor(s1, off, 32);
          s2 += __shfl_xor(s2, off, 32);
        }
        if (l == 0) { atomicAdd(&lsum[o], s1); atomicAdd(&lsq[o], s2); }
      }
    }
  }
  if (!WRITE_OUT) {
    __syncthreads();
    if (tid < 64) { atomicAdd(&gsum[tid], lsum[tid]); atomicAdd(&gsq[tid], lsq[tid]); }
  }
}

__global__ void k_bnstats(const float* __restrict__ gsum, const float* __restrict__ gsq,
                          const float* __restrict__ gamma, const float* __restrict__ beta,
                          float* __restrict__ ss)
{
  const int ch = threadIdx.x;
  if (ch < 64) {
    const float cnt  = (float)((size_t)NB * HWD);
    const float mean = gsum[ch] / cnt;
    const float var  = gsq[ch] / cnt - mean * mean;
    const float sc   = gamma[ch] * rsqrtf(var + 1e-5f);
    ss[ch]      = sc;
    ss[64 + ch] = beta[ch] - mean * sc;
  }
}

__global__ void k_zero(float* __restrict__ p) { p[threadIdx.x] = 0.f; }

// ---------------------------------------------------------------------------
extern "C" void kernel_launch(void* const* d_in, const int* in_sizes, int n_in,
                              void* d_out, int out_size, void* d_ws, size_t ws_size,
                              hipStream_t stream)
{
  const float* seg   = (const float*)d_in[0];
  const float* depth = (const float*)d_in[1];
  const float* WG    = (const float*)d_in[2];
  const float* bG    = (const float*)d_in[3];
  const float* WF    = (const float*)d_in[4];
  const float* bF    = (const float*)d_in[5];
  const float* WK    = (const float*)d_in[6];
  const float* bK    = (const float*)d_in[7];
  const float* gamma = (const float*)d_in[8];
  const float* beta  = (const float*)d_in[9];
  float* out = (float*)d_out;

  char* ws = (char*)d_ws;
  float* outF = (float*)ws;                                   // 8*64*65536 f32
  float* diag = (float*)(ws + (size_t)NB * C64 * HWD * 4);    // 8*65536 f32
  float* gsum = (float*)(ws + (size_t)NB * C64 * HWD * 4 + (size_t)NB * HWD * 4);
  float* gsq  = gsum + 64;
  float* ss   = gsum + 128;                                   // scale+shift

  hipLaunchKernelGGL(k_zero, dim3(1), dim3(128), 0, stream, gsum);
  hipLaunchKernelGGL(k_convF, dim3(512, NB), dim3(256), 0, stream,
                     seg, WF, bF, outF);
  hipLaunchKernelGGL(k_diag, dim3(1024, NB), dim3(256), 0, stream,
                     seg, WK, bK, outF, diag);
  hipLaunchKernelGGL((k_convG<false>), dim3(512, NB), dim3(256), 0, stream,
                     depth, WG, bG, diag, (const float*)nullptr, gsum, gsq, out);
  hipLaunchKernelGGL(k_bnstats, dim3(1), dim3(64), 0, stream,
                     gsum, gsq, gamma, beta, ss);
  hipLaunchKernelGGL((k_convG<true>), dim3(512, NB), dim3(256), 0, stream,
                     depth, WG, bG, diag, ss, (float*)nullptr, (float*)nullptr, out);
}